// WindowAttention_59811714564810
// MI455X (gfx1250) — compile-verified
//
#include <hip/hip_runtime.h>
#include <math.h>

typedef __attribute__((ext_vector_type(2))) float v2f;
typedef __attribute__((ext_vector_type(8))) float v8f;

#define B_ 16
#define T_ 12
#define N_ 512
#define C_ 64
#define KH_ 8
#define HS_ 8
#define CUTS_ 4
#define CS_ 3
#define P_ 2
#define BN_ (B_ * N_)          // 8192
#define MROWS_ (B_ * P_ * N_)  // 16384

__device__ __forceinline__ v8f wmma4(v2f a, v2f b, v8f c) {
  return __builtin_amdgcn_wmma_f32_16x16x4_f32(false, a, false, b, (short)0, c,
                                               false, false);
}

// ---------------------------------------------------------------------------
// xm = mean over T of x  :  (b, T, n, c) -> (b*n*c)
// ---------------------------------------------------------------------------
__global__ void mean_kernel(const float* __restrict__ x, float* __restrict__ xm) {
  int idx = blockIdx.x * 256 + threadIdx.x;          // < B*N*C = 524288
  int c = idx & 63;
  int n = (idx >> 6) & 511;
  int b = idx >> 15;
  float s = 0.f;
#pragma unroll
  for (int t = 0; t < T_; ++t)
    s += x[(((size_t)b * T_ + t) * N_ + n) * C_ + c];
  xm[idx] = s * (1.f / T_);
}

// ---------------------------------------------------------------------------
// Per (b,n): h2 for the 4 "wg" generator groups (store 5 floats each) and
// bias bi for the 4 "bg" groups (store 64 floats each).
// ---------------------------------------------------------------------------
__global__ void pg_kernel(const float* __restrict__ xm,
                          const float* __restrict__ wgw1, const float* __restrict__ wgb1,
                          const float* __restrict__ wgw2, const float* __restrict__ wgb2,
                          const float* __restrict__ bgw1, const float* __restrict__ bgb1,
                          const float* __restrict__ bgw2, const float* __restrict__ bgb2,
                          const float* __restrict__ bgw3, const float* __restrict__ bgb3,
                          float* __restrict__ h2out, float* __restrict__ biout) {
  int bn = blockIdx.x;       // 8192
  int tid = threadIdx.x;     // 64
  __shared__ float xr[64];
  __shared__ float h1[32];
  __shared__ float h2s[5];

  xr[tid] = xm[(size_t)bn * 64 + tid];
  __syncthreads();

  for (int g = 0; g < 4; ++g) {
    // ---- wg path: store h2 ----
    if (tid < 32) {
      float s = wgb1[g * 32 + tid];
      for (int c = 0; c < 64; ++c) s += xr[c] * wgw1[((size_t)g * 64 + c) * 32 + tid];
      h1[tid] = fmaxf(s, 0.f);
    }
    __syncthreads();
    if (tid < 5) {
      float s = wgb2[g * 5 + tid];
      for (int k = 0; k < 32; ++k) s += h1[k] * wgw2[((size_t)g * 32 + k) * 5 + tid];
      h2out[((size_t)g * BN_ + bn) * 5 + tid] = fmaxf(s, 0.f);
    }
    __syncthreads();
    // ---- bg path: compute bias ----
    if (tid < 32) {
      float s = bgb1[g * 32 + tid];
      for (int c = 0; c < 64; ++c) s += xr[c] * bgw1[((size_t)g * 64 + c) * 32 + tid];
      h1[tid] = fmaxf(s, 0.f);
    }
    __syncthreads();
    if (tid < 5) {
      float s = bgb2[g * 5 + tid];
      for (int k = 0; k < 32; ++k) s += h1[k] * bgw2[((size_t)g * 32 + k) * 5 + tid];
      h2s[tid] = fmaxf(s, 0.f);
    }
    __syncthreads();
    {
      float s = bgb3[g * 64 + tid];
#pragma unroll
      for (int j = 0; j < 5; ++j) s += h2s[j] * bgw3[((size_t)g * 5 + j) * 64 + tid];
      biout[((size_t)g * BN_ + bn) * 64 + tid] = s;
    }
    __syncthreads();
  }
}

// ---------------------------------------------------------------------------
// Fused per-(b,n): build tcat, rebuild per-position 64x64 weights for groups
// 0/1 in LDS, apply (5-token lincustom) to get k,v, then attention-1 over the
// 5 tokens (proxies as queries). Output o : (b, P, n, c).
// ---------------------------------------------------------------------------
__global__ void cutA_kernel(const float* __restrict__ x, const float* __restrict__ proxies,
                            const float* __restrict__ out_prev,
                            const float* __restrict__ h2, const float* __restrict__ bi,
                            const float* __restrict__ w3, const float* __restrict__ b3,
                            float* __restrict__ obuf, int cut, int use_prev) {
  int bn = blockIdx.x;   // 8192
  int b = bn >> 9, n = bn & 511;
  int tid = threadIdx.x; // 256

  __shared__ float tc[5 * 64];
  __shared__ float wl[64 * 65];
  __shared__ float kl[5 * 64];
  __shared__ float vl[5 * 64];
  __shared__ float h2s[2][5];

  for (int idx = tid; idx < 5 * 64; idx += 256) {
    int t = idx >> 6, c = idx & 63;
    float v;
    if (t < P_) {
      v = proxies[(((size_t)(cut * P_ + t)) * N_ + n) * 64 + c];
      if (use_prev) v += out_prev[((size_t)b * N_ + n) * 64 + c];
    } else {
      v = x[(((size_t)b * T_ + (cut * CS_ + t - P_)) * N_ + n) * 64 + c];
    }
    tc[idx] = v;
  }
  if (tid < 10) {
    int g = tid / 5, j = tid % 5;
    h2s[g][j] = h2[((size_t)g * BN_ + bn) * 5 + j];
  }
  __syncthreads();

  for (int g = 0; g < 2; ++g) {
    float hh[5];
#pragma unroll
    for (int j = 0; j < 5; ++j) hh[j] = h2s[g][j];
    const float* w3g = w3 + (size_t)g * 5 * 4096;
    const float* b3g = b3 + (size_t)g * 4096;
    for (int idx = tid; idx < 4096; idx += 256) {
      float s = b3g[idx];
#pragma unroll
      for (int j = 0; j < 5; ++j) s += hh[j] * w3g[(size_t)j * 4096 + idx];
      int c = idx >> 6, d = idx & 63;
      wl[c * 65 + d] = s;
    }
    __syncthreads();
    const float* big = bi + ((size_t)g * BN_ + bn) * 64;
    float* dst = (g == 0) ? kl : vl;
    for (int idx = tid; idx < 5 * 64; idx += 256) {
      int t = idx >> 6, d = idx & 63;
      float s = big[d];
      for (int c = 0; c < 64; ++c) s += tc[t * 64 + c] * wl[c * 65 + d];
      dst[idx] = s;
    }
    __syncthreads();
  }

  // attention-1: softmax over the 5 tokens, per (p, head, h)
  if (tid < 128) {
    int p = tid >> 6, kh = (tid >> 3) & 7, h = tid & 7;
    const float iscale = 0.35355339059327373f;  // 1/sqrt(8)
    float q[8];
#pragma unroll
    for (int j = 0; j < 8; ++j) q[j] = tc[p * 64 + kh * 8 + j];
    float sc[5];
    float mx = -INFINITY;
#pragma unroll
    for (int t = 0; t < 5; ++t) {
      float s = 0.f;
#pragma unroll
      for (int j = 0; j < 8; ++j) s += q[j] * kl[t * 64 + kh * 8 + j];
      s *= iscale;
      sc[t] = s;
      mx = fmaxf(mx, s);
    }
    float sum = 0.f;
#pragma unroll
    for (int t = 0; t < 5; ++t) { sc[t] = __expf(sc[t] - mx); sum += sc[t]; }
    float acc = 0.f;
#pragma unroll
    for (int t = 0; t < 5; ++t) acc += sc[t] * vl[t * 64 + kh * 8 + h];
    obuf[(((size_t)b * P_ + p) * N_ + n) * 64 + kh * 8 + h] = acc / sum;
  }
}

// ---------------------------------------------------------------------------
// Generic WMMA fp32 GEMM:  OUT[M x 64] = act( IN[M x 64] @ W[64 x 64] + bias )
// 128 threads = 4 waves; wave w computes a 16x16 tile at cols w*16.
// act: 0 none, 1 tanh, 2 relu, 3 sigmoid
// ---------------------------------------------------------------------------
__global__ void gemm64_kernel(const float* __restrict__ IN, const float* __restrict__ W,
                              const float* __restrict__ bias, float* __restrict__ OUT,
                              int act) {
  int wave = threadIdx.x >> 5;
  int lane = threadIdx.x & 31;
  int m0 = blockIdx.x * 16;
  int n0 = wave * 16;
  int mrow = lane & 15;
  int half = lane >> 4;
  int ncol = lane & 15;

  v8f acc;
  float bv = bias[n0 + ncol];
#pragma unroll
  for (int r = 0; r < 8; ++r) acc[r] = bv;

#pragma unroll
  for (int k0 = 0; k0 < 64; k0 += 4) {
    int kb = k0 + half * 2;
    v2f a, bb;
    a.x = IN[((size_t)(m0 + mrow)) * 64 + kb];
    a.y = IN[((size_t)(m0 + mrow)) * 64 + kb + 1];
    bb.x = W[(size_t)kb * 64 + n0 + ncol];
    bb.y = W[(size_t)(kb + 1) * 64 + n0 + ncol];
    acc = wmma4(a, bb, acc);
  }

#pragma unroll
  for (int r = 0; r < 8; ++r) {
    float v = acc[r];
    if (act == 1) v = tanhf(v);
    else if (act == 2) v = fmaxf(v, 0.f);
    else if (act == 3) v = 1.f / (1.f + __expf(-v));
    int md = r + half * 8;
    OUT[((size_t)(m0 + md)) * 64 + n0 + ncol] = v;
  }
}

// ---------------------------------------------------------------------------
// Per (b,n): rebuild 64x64 weights for groups 2/3, apply to the 2-token o
// (post t-MLP) producing ks, vs : (b, P, n, c).
// ---------------------------------------------------------------------------
__global__ void cutS_kernel(const float* __restrict__ oo, const float* __restrict__ h2,
                            const float* __restrict__ bi, const float* __restrict__ w3,
                            const float* __restrict__ b3, float* __restrict__ ksb,
                            float* __restrict__ vsb) {
  int bn = blockIdx.x;
  int b = bn >> 9, n = bn & 511;
  int tid = threadIdx.x;  // 256

  __shared__ float tc[2 * 64];
  __shared__ float wl[64 * 65];
  __shared__ float h2s[2][5];

  if (tid < 128) {
    int p = tid >> 6, c = tid & 63;
    tc[tid] = oo[(((size_t)b * P_ + p) * N_ + n) * 64 + c];
  }
  if (tid < 10) {
    int g = tid / 5, j = tid % 5;
    h2s[g][j] = h2[((size_t)(g + 2) * BN_ + bn) * 5 + j];
  }
  __syncthreads();

  for (int gg = 0; gg < 2; ++gg) {
    int g = gg + 2;
    float hh[5];
#pragma unroll
    for (int j = 0; j < 5; ++j) hh[j] = h2s[gg][j];
    const float* w3g = w3 + (size_t)g * 5 * 4096;
    const float* b3g = b3 + (size_t)g * 4096;
    for (int idx = tid; idx < 4096; idx += 256) {
      float s = b3g[idx];
#pragma unroll
      for (int j = 0; j < 5; ++j) s += hh[j] * w3g[(size_t)j * 4096 + idx];
      int c = idx >> 6, d = idx & 63;
      wl[c * 65 + d] = s;
    }
    __syncthreads();
    const float* big = bi + ((size_t)g * BN_ + bn) * 64;
    float* dst = (gg == 0) ? ksb : vsb;
    for (int idx = tid; idx < 128; idx += 256) {
      int t = idx >> 6, d = idx & 63;
      float s = big[d];
      for (int c = 0; c < 64; ++c) s += tc[t * 64 + c] * wl[c * 65 + d];
      dst[(((size_t)b * P_ + t) * N_ + n) * 64 + d] = s;
    }
    __syncthreads();
  }
}

// ---------------------------------------------------------------------------
// Attention-2: per (b, p, head) flash attention over n=512 tokens, hs=8.
// 256 threads = 8 waves; wave owns 64 query rows (4 row-tiles of 16).
// S tiles and P.V via V_WMMA_F32_16X16X4_F32; online softmax with per-wave
// LDS staging (also re-layouts P from D-layout to A-layout).
// ---------------------------------------------------------------------------
__global__ void attn2_kernel(const float* __restrict__ oo, const float* __restrict__ ks,
                             const float* __restrict__ vs, float* __restrict__ o2out) {
  int blk = blockIdx.x;  // 256 = b*p*kh
  int b = blk >> 4;
  int p = (blk >> 3) & 1;
  int kh = blk & 7;
  int tid = threadIdx.x;  // 256
  int wave = tid >> 5, lane = tid & 31;
  int half = lane >> 4, l15 = lane & 15;

  __shared__ float Kl[512 * 8];
  __shared__ float Vl[512 * 8];
  __shared__ float Pst[8][16 * 17];
  __shared__ float Alp[8][16];
  __shared__ float Lrow[8][16];

  const size_t basebp = ((size_t)(b * P_ + p)) * N_ * 64 + (size_t)kh * 8;

  for (int idx = tid; idx < 512 * 8; idx += 256) {
    int m = idx >> 3, h = idx & 7;
    Kl[idx] = ks[basebp + (size_t)m * 64 + h];
    Vl[idx] = vs[basebp + (size_t)m * 64 + h];
  }
  __syncthreads();

  const float iscale = 0.35355339059327373f;  // 1/sqrt(8)

  for (int rt = 0; rt < 4; ++rt) {
    int r0 = wave * 64 + rt * 16;
    // Q tile in A-layout registers (two K=4 steps cover hs=8)
    v2f qa0, qa1;
    {
      const float* qp = oo + basebp + (size_t)(r0 + l15) * 64;
      int kb = half * 2;
      qa0.x = qp[kb];     qa0.y = qp[kb + 1];
      qa1.x = qp[4 + kb]; qa1.y = qp[4 + kb + 1];
    }
    float mrun = -INFINITY, lrun = 0.f;
    v8f oacc = {};

    for (int ct = 0; ct < 32; ++ct) {
      int m0 = ct * 16;
      v8f s = {};
      {
        int kb = half * 2;
        v2f bb;
        bb.x = Kl[(m0 + l15) * 8 + kb];
        bb.y = Kl[(m0 + l15) * 8 + kb + 1];
        s = wmma4(qa0, bb, s);
        bb.x = Kl[(m0 + l15) * 8 + 4 + kb];
        bb.y = Kl[(m0 + l15) * 8 + 4 + kb + 1];
        s = wmma4(qa1, bb, s);
      }
#pragma unroll
      for (int r = 0; r < 8; ++r)
        Pst[wave][(r + half * 8) * 17 + l15] = s[r] * iscale;
      __syncthreads();

      if (lane < 16) {
        float* row = &Pst[wave][lane * 17];
        float tmax = row[0];
#pragma unroll
        for (int j = 1; j < 16; ++j) tmax = fmaxf(tmax, row[j]);
        float mnew = fmaxf(mrun, tmax);
        float alpha = __expf(mrun - mnew);
        float sum = 0.f;
#pragma unroll
        for (int j = 0; j < 16; ++j) {
          float e = __expf(row[j] - mnew);
          row[j] = e;
          sum += e;
        }
        lrun = lrun * alpha + sum;
        mrun = mnew;
        Alp[wave][lane] = alpha;
      }
      __syncthreads();

#pragma unroll
      for (int r = 0; r < 8; ++r) oacc[r] *= Alp[wave][r + half * 8];

#pragma unroll
      for (int ks4 = 0; ks4 < 16; ks4 += 4) {
        int kb = ks4 + half * 2;
        v2f a, bb;
        a.x = Pst[wave][l15 * 17 + kb];
        a.y = Pst[wave][l15 * 17 + kb + 1];
        bb.x = (l15 < 8) ? Vl[(m0 + kb) * 8 + l15] : 0.f;
        bb.y = (l15 < 8) ? Vl[(m0 + kb + 1) * 8 + l15] : 0.f;
        oacc = wmma4(a, bb, oacc);
      }
    }

    if (lane < 16) Lrow[wave][lane] = lrun;
    __syncthreads();
#pragma unroll
    for (int r = 0; r < 8; ++r) {
      int md = r + half * 8;
      float v = oacc[r] / Lrow[wave][md];
      if (l15 < 8)
        o2out[basebp + (size_t)(r0 + md) * 64 + l15] = v;
    }
    __syncthreads();
  }
}

// ---------------------------------------------------------------------------
// out = sum_p gate * o2 ; write running out_prev and the chunk of d_out.
// ---------------------------------------------------------------------------
__global__ void agg_kernel(const float* __restrict__ gate, const float* __restrict__ oo2,
                           float* __restrict__ out_prev, float* __restrict__ dout,
                           int cut) {
  int idx = blockIdx.x * 256 + threadIdx.x;  // < B*N*C
  int c = idx & 63;
  int n = (idx >> 6) & 511;
  int b = idx >> 15;
  size_t i0 = (((size_t)b * P_ + 0) * N_ + n) * 64 + c;
  size_t i1 = (((size_t)b * P_ + 1) * N_ + n) * 64 + c;
  float v = gate[i0] * oo2[i0] + gate[i1] * oo2[i1];
  out_prev[idx] = v;
  dout[(((size_t)b * CUTS_ + cut) * N_ + n) * 64 + c] = v;
}

// ---------------------------------------------------------------------------
extern "C" void kernel_launch(void* const* d_in, const int* in_sizes, int n_in,
                              void* d_out, int out_size, void* d_ws, size_t ws_size,
                              hipStream_t stream) {
  const float* x       = (const float*)d_in[0];
  const float* proxies = (const float*)d_in[1];
  const float* wgw1 = (const float*)d_in[2];
  const float* wgb1 = (const float*)d_in[3];
  const float* wgw2 = (const float*)d_in[4];
  const float* wgb2 = (const float*)d_in[5];
  const float* wgw3 = (const float*)d_in[6];
  const float* wgb3 = (const float*)d_in[7];
  const float* bgw1 = (const float*)d_in[8];
  const float* bgb1 = (const float*)d_in[9];
  const float* bgw2 = (const float*)d_in[10];
  const float* bgb2 = (const float*)d_in[11];
  const float* bgw3 = (const float*)d_in[12];
  const float* bgb3 = (const float*)d_in[13];
  const float* tp1w = (const float*)d_in[14];
  const float* tp1b = (const float*)d_in[15];
  const float* tp2w = (const float*)d_in[16];
  const float* tp2b = (const float*)d_in[17];
  const float* sp1w = (const float*)d_in[18];
  const float* sp1b = (const float*)d_in[19];
  const float* sp2w = (const float*)d_in[20];
  const float* sp2b = (const float*)d_in[21];
  const float* aw1  = (const float*)d_in[22];
  const float* ab1  = (const float*)d_in[23];
  const float* aw2  = (const float*)d_in[24];
  const float* ab2  = (const float*)d_in[25];

  float* ws = (float*)d_ws;
  float* xm       = ws;                      // 524288
  float* h2       = xm + 524288;             // 163840
  float* bi       = h2 + 163840;             // 2097152
  float* out_prev = bi + 2097152;            // 524288
  float* obuf     = out_prev + 524288;       // 1048576
  float* tmp      = obuf + 1048576;          // 1048576
  float* oo       = tmp + 1048576;           // 1048576
  float* ksb      = oo + 1048576;            // 1048576
  float* vsb      = ksb + 1048576;           // 1048576
  float* o2b      = vsb + 1048576;           // 1048576
  float* oo2      = o2b + 1048576;           // 1048576
  float* gateb    = oo2 + 1048576;           // 1048576  (total ~46.8 MB)

  mean_kernel<<<2048, 256, 0, stream>>>(x, xm);
  pg_kernel<<<BN_, 64, 0, stream>>>(xm, wgw1, wgb1, wgw2, wgb2,
                                    bgw1, bgb1, bgw2, bgb2, bgw3, bgb3, h2, bi);

  for (int cut = 0; cut < CUTS_; ++cut) {
    cutA_kernel<<<BN_, 256, 0, stream>>>(x, proxies, out_prev, h2, bi, wgw3, wgb3,
                                         obuf, cut, cut > 0 ? 1 : 0);
    gemm64_kernel<<<MROWS_ / 16, 128, 0, stream>>>(obuf, tp1w, tp1b, tmp, 1);
    gemm64_kernel<<<MROWS_ / 16, 128, 0, stream>>>(tmp, tp2w, tp2b, oo, 0);
    cutS_kernel<<<BN_, 256, 0, stream>>>(oo, h2, bi, wgw3, wgb3, ksb, vsb);
    attn2_kernel<<<B_ * P_ * KH_, 256, 0, stream>>>(oo, ksb, vsb, o2b);
    gemm64_kernel<<<MROWS_ / 16, 128, 0, stream>>>(o2b, sp1w, sp1b, tmp, 2);
    gemm64_kernel<<<MROWS_ / 16, 128, 0, stream>>>(tmp, sp2w, sp2b, oo2, 0);
    gemm64_kernel<<<MROWS_ / 16, 128, 0, stream>>>(oo2, aw1, ab1, tmp, 2);
    gemm64_kernel<<<MROWS_ / 16, 128, 0, stream>>>(tmp, aw2, ab2, gateb, 3);
    agg_kernel<<<2048, 256, 0, stream>>>(gateb, oo2, out_prev, (float*)d_out, cut);
  }
}